// ScalarSGC_44624710205618
// MI455X (gfx1250) — compile-verified
//
#include <hip/hip_runtime.h>
#include <stdint.h>

#define D_FEAT 128   // features per node
#define WAVES_PER_BLOCK 8

typedef unsigned int u32;

static __device__ __forceinline__ u32 umin32(u32 a, u32 b) { return a < b ? a : b; }

// ---------------------------------------------------------------------------
// CDNA5 async global->LDS copy (ASYNCcnt-tracked) with builtin/asm fallback
// Builtin signature (probe-confirmed): (v2i AS1*, v2i AS3*, Ii offset, Ii cpol)
// ---------------------------------------------------------------------------
typedef int v2i __attribute__((vector_size(2 * sizeof(int))));
typedef __attribute__((address_space(1))) v2i* gptr_v2i;
typedef __attribute__((address_space(3))) v2i* lptr_v2i;

static __device__ __forceinline__ void async_load_pair(const uint2* gsrc, uint2* lds_dst) {
#if __has_builtin(__builtin_amdgcn_global_load_async_to_lds_b64)
  __builtin_amdgcn_global_load_async_to_lds_b64(
      (gptr_v2i)(uintptr_t)gsrc,
      (lptr_v2i)(uint32_t)(uintptr_t)lds_dst,
      0, 0);
#else
  asm volatile("global_load_async_to_lds_b64 %0, %1, off"
               :: "v"((uint32_t)(uintptr_t)lds_dst), "v"(gsrc)
               : "memory");
#endif
}

static __device__ __forceinline__ void wait_async_le(int n) {
#if __has_builtin(__builtin_amdgcn_s_wait_asynccnt)
  if (n == 0) __builtin_amdgcn_s_wait_asynccnt(0);
  else        __builtin_amdgcn_s_wait_asynccnt(1);
#else
  if (n == 0) asm volatile("s_wait_asynccnt 0" ::: "memory");
  else        asm volatile("s_wait_asynccnt 1" ::: "memory");
#endif
}

// ---------------------------------------------------------------------------
// Small utility kernels
// ---------------------------------------------------------------------------
__global__ void zero_u32_kernel(u32* p, long n) {
  long i = (long)blockIdx.x * blockDim.x + threadIdx.x;
  long stride = (long)gridDim.x * blockDim.x;
  for (; i < n; i += stride) p[i] = 0u;
}

__global__ void hist_kernel(const int* __restrict__ rows, u32* __restrict__ counts, int E) {
  long i = (long)blockIdx.x * blockDim.x + threadIdx.x;
  long stride = (long)gridDim.x * blockDim.x;
  for (; i < E; i += stride) atomicAdd(&counts[rows[i]], 1u);
}

// Single-block serial-tile scan using wave32 shuffle scans (4 barriers/tile).
// Writes offsets[0..n] (inclusive prefix shifted to [i+1]) and
// cursor[i] = exclusive prefix (write cursors for the fill pass).
__global__ __launch_bounds__(1024) void scan_kernel(const u32* __restrict__ counts,
                                                    u32* __restrict__ offsets,
                                                    u32* __restrict__ cursor, int n) {
  __shared__ u32 wsum[32];   // 1024 threads = 32 wave32s
  __shared__ u32 carry_s;
  const int tid = threadIdx.x;
  const int lane = tid & 31;
  const int wid = tid >> 5;
  if (tid == 0) { carry_s = 0u; offsets[0] = 0u; }
  __syncthreads();
  for (int base = 0; base < n; base += 1024) {
    const int i = base + tid;
    const u32 v = (i < n) ? counts[i] : 0u;
    // intra-wave inclusive scan (5 shfl steps, no barriers)
    u32 s = v;
    #pragma unroll
    for (int d = 1; d < 32; d <<= 1) {
      u32 t = __shfl_up(s, d, 32);
      if (lane >= d) s += t;
    }
    if (lane == 31) wsum[wid] = s;
    __syncthreads();
    if (wid == 0) {        // wave 0 scans the 32 wave totals
      u32 ws = wsum[lane];
      #pragma unroll
      for (int d = 1; d < 32; d <<= 1) {
        u32 t = __shfl_up(ws, d, 32);
        if (lane >= d) ws += t;
      }
      wsum[lane] = ws;     // inclusive prefix of wave totals
    }
    __syncthreads();
    const u32 waveoff = (wid == 0) ? 0u : wsum[wid - 1];
    const u32 inc = s + waveoff + carry_s;   // inclusive prefix incl. carry
    if (i < n) { offsets[i + 1] = inc; cursor[i] = inc - v; }
    __syncthreads();                          // everyone has read carry_s / wsum
    if (tid == 1023) carry_s = inc;           // tile total + old carry
    __syncthreads();
  }
}

// Scatter edges into CSR order as packed (col, weight_bits) pairs.
__global__ void fill_kernel(const int* __restrict__ rows, const int* __restrict__ cols,
                            const float* __restrict__ vals, u32* __restrict__ cursor,
                            uint2* __restrict__ epair, int E) {
  long i = (long)blockIdx.x * blockDim.x + threadIdx.x;
  long stride = (long)gridDim.x * blockDim.x;
  for (; i < E; i += stride) {
    const int r = rows[i];
    const u32 pos = atomicAdd(&cursor[r], 1u);
    epair[pos] = make_uint2((u32)cols[i], __float_as_uint(vals[i]));
  }
}

// ---------------------------------------------------------------------------
// Main layer kernel (CSR pull): one wave32 per destination node.
// Lane l owns float4 features [4l..4l+3] (32 lanes * 4 = 128 = D_FEAT).
// Edge (col,weight) chunks are double-buffered into LDS with async loads.
// Inner loop unrolled x4 (4 independent b128 gathers in flight per wave).
// ELU + layer scalar fused into the epilogue.
// ---------------------------------------------------------------------------
__global__ __launch_bounds__(256) void sgc_layer_csr(
    const float* __restrict__ h_in, const u32* __restrict__ offsets,
    const uint2* __restrict__ epair, const float* __restrict__ scalars,
    int layer, float* __restrict__ h_out, int n_nodes) {
  __shared__ uint2 stage[WAVES_PER_BLOCK][2][32];
  const int wid = threadIdx.x >> 5;
  const int lane = threadIdx.x & 31;
  const int node = blockIdx.x * WAVES_PER_BLOCK + wid;
  if (node >= n_nodes) return;

  const u32 start = offsets[node];
  const u32 cnt = offsets[node + 1] - start;
  float4 acc = make_float4(0.f, 0.f, 0.f, 0.f);

  const u32 nch = (cnt + 31u) >> 5;
  if (nch) {
    // prefetch chunk 0
    if ((u32)lane < umin32(cnt, 32u))
      async_load_pair(epair + start + (u32)lane, &stage[wid][0][lane]);
    for (u32 c = 0; c < nch; ++c) {
      if (c + 1u < nch) {
        const u32 cn = umin32(cnt - (c + 1u) * 32u, 32u);
        if ((u32)lane < cn)
          async_load_pair(epair + start + (c + 1u) * 32u + (u32)lane,
                          &stage[wid][(c + 1u) & 1u][lane]);
        wait_async_le(1);   // chunk c done (async loads complete in order)
      } else {
        wait_async_le(0);
      }
      const u32 here = umin32(cnt - c * 32u, 32u);
      const uint2* buf = &stage[wid][c & 1u][0];
      u32 j = 0;
      // x4 unroll: two 16B LDS reads, four independent global b128 gathers
      for (; j + 4u <= here; j += 4u) {
        const uint4* b4 = (const uint4*)(buf + j);   // 16B-aligned (j % 4 == 0)
        const uint4 q0 = b4[0];
        const uint4 q1 = b4[1];
        const float w0 = __uint_as_float(q0.y);
        const float w1 = __uint_as_float(q0.w);
        const float w2 = __uint_as_float(q1.y);
        const float w3 = __uint_as_float(q1.w);
        const float4 v0 = ((const float4*)(h_in + (size_t)q0.x * D_FEAT))[lane];
        const float4 v1 = ((const float4*)(h_in + (size_t)q0.z * D_FEAT))[lane];
        const float4 v2 = ((const float4*)(h_in + (size_t)q1.x * D_FEAT))[lane];
        const float4 v3 = ((const float4*)(h_in + (size_t)q1.z * D_FEAT))[lane];
        acc.x += w0 * v0.x; acc.y += w0 * v0.y; acc.z += w0 * v0.z; acc.w += w0 * v0.w;
        acc.x += w1 * v1.x; acc.y += w1 * v1.y; acc.z += w1 * v1.z; acc.w += w1 * v1.w;
        acc.x += w2 * v2.x; acc.y += w2 * v2.y; acc.z += w2 * v2.z; acc.w += w2 * v2.w;
        acc.x += w3 * v3.x; acc.y += w3 * v3.y; acc.z += w3 * v3.z; acc.w += w3 * v3.w;
      }
      for (; j < here; ++j) {
        const uint2 p = buf[j];
        const float w = __uint_as_float(p.y);
        const float4 v = ((const float4*)(h_in + (size_t)p.x * D_FEAT))[lane];
        acc.x += w * v.x; acc.y += w * v.y; acc.z += w * v.z; acc.w += w * v.w;
      }
    }
  }

  const float s = scalars[layer];
  acc.x *= s; acc.y *= s; acc.z *= s; acc.w *= s;
  acc.x = acc.x > 0.f ? acc.x : __expf(acc.x) - 1.f;
  acc.y = acc.y > 0.f ? acc.y : __expf(acc.y) - 1.f;
  acc.z = acc.z > 0.f ? acc.z : __expf(acc.z) - 1.f;
  acc.w = acc.w > 0.f ? acc.w : __expf(acc.w) - 1.f;
  ((float4*)(h_out + (size_t)node * D_FEAT))[lane] = acc;
}

// ---------------------------------------------------------------------------
// Fallback path (small workspace): wave-per-edge atomic scatter + ELU pass.
// ---------------------------------------------------------------------------
__global__ void edge_scatter_atomic(const int* __restrict__ rows, const int* __restrict__ cols,
                                    const float* __restrict__ vals,
                                    const float* __restrict__ scalars, int layer,
                                    const float* __restrict__ h_in, float* __restrict__ agg,
                                    int E) {
  const int lane = threadIdx.x & 31;
  long w0 = (long)blockIdx.x * (blockDim.x >> 5) + (threadIdx.x >> 5);
  long nw = (long)gridDim.x * (blockDim.x >> 5);
  const float s = scalars[layer];
  for (long e = w0; e < E; e += nw) {
    const int r = rows[e];
    const int c = cols[e];
    const float w = vals[e] * s;
    const float4 v = ((const float4*)(h_in + (size_t)c * D_FEAT))[lane];
    float* dst = agg + (size_t)r * D_FEAT + lane * 4;
    __hip_atomic_fetch_add(dst + 0, w * v.x, __ATOMIC_RELAXED, __HIP_MEMORY_SCOPE_AGENT);
    __hip_atomic_fetch_add(dst + 1, w * v.y, __ATOMIC_RELAXED, __HIP_MEMORY_SCOPE_AGENT);
    __hip_atomic_fetch_add(dst + 2, w * v.z, __ATOMIC_RELAXED, __HIP_MEMORY_SCOPE_AGENT);
    __hip_atomic_fetch_add(dst + 3, w * v.w, __ATOMIC_RELAXED, __HIP_MEMORY_SCOPE_AGENT);
  }
}

__global__ void elu_inplace(float* p, long n) {
  long i = (long)blockIdx.x * blockDim.x + threadIdx.x;
  long stride = (long)gridDim.x * blockDim.x;
  for (; i < n; i += stride) {
    const float v = p[i];
    p[i] = v > 0.f ? v : __expf(v) - 1.f;
  }
}

// ---------------------------------------------------------------------------
// Launch
// ---------------------------------------------------------------------------
extern "C" void kernel_launch(void* const* d_in, const int* in_sizes, int n_in,
                              void* d_out, int out_size, void* d_ws, size_t ws_size,
                              hipStream_t stream) {
  const float* x      = (const float*)d_in[0];
  const int*   erow   = (const int*)d_in[1];
  const int*   ecol   = (const int*)d_in[2];
  const float* evals  = (const float*)d_in[3];
  const float* scalrs = (const float*)d_in[4];
  const int N = in_sizes[0] / D_FEAT;
  const int E = in_sizes[1];
  float* out = (float*)d_out;

  auto align256 = [](size_t v) { return (v + 255) & ~(size_t)255; };
  const size_t off_b = align256((size_t)(N + 1) * 4);
  const size_t cur_b = align256((size_t)N * 4);
  const size_t h1_b  = align256((size_t)N * D_FEAT * 4);
  const size_t ep_b  = align256((size_t)E * 8);
  const size_t need  = off_b + cur_b + h1_b + ep_b;

  char* ws = (char*)d_ws;
  const int eb = (E + 255) / 256;                  // edge-parallel grid
  const int nb = (N + 255) / 256;
  const long ND = (long)N * D_FEAT;
  const int ndb = (int)((ND + 255) / 256);

  if (ws_size >= need) {
    u32*   offsets = (u32*)ws;
    u32*   cursor  = (u32*)(ws + off_b);
    float* h1      = (float*)(ws + off_b + cur_b);
    uint2* epair   = (uint2*)(ws + off_b + cur_b + h1_b);
    u32*   counts  = (u32*)epair;  // aliased: counts are dead before fill writes epair

    // Build CSR (same adjacency both layers; rebuilt every call -> deterministic work)
    zero_u32_kernel<<<nb, 256, 0, stream>>>(counts, N);
    hist_kernel<<<eb, 256, 0, stream>>>(erow, counts, E);
    scan_kernel<<<1, 1024, 0, stream>>>(counts, offsets, cursor, N);
    fill_kernel<<<eb, 256, 0, stream>>>(erow, ecol, evals, cursor, epair, E);

    const int lb = (N + WAVES_PER_BLOCK - 1) / WAVES_PER_BLOCK;
    sgc_layer_csr<<<lb, 256, 0, stream>>>(x,  offsets, epair, scalrs, 0, h1,  N);
    sgc_layer_csr<<<lb, 256, 0, stream>>>(h1, offsets, epair, scalrs, 1, out, N);
  } else if (ws_size >= h1_b) {
    float* h1 = (float*)ws;
    zero_u32_kernel<<<ndb, 256, 0, stream>>>((u32*)h1, ND);
    { const int gb = (E + 7) / 8;  // 8 waves per 256-thread block, one wave per edge
      edge_scatter_atomic<<<gb, 256, 0, stream>>>(erow, ecol, evals, scalrs, 0, x, h1, E);
      elu_inplace<<<ndb, 256, 0, stream>>>(h1, ND);
      zero_u32_kernel<<<ndb, 256, 0, stream>>>((u32*)out, ND);
      edge_scatter_atomic<<<gb, 256, 0, stream>>>(erow, ecol, evals, scalrs, 1, h1, out, E);
      elu_inplace<<<ndb, 256, 0, stream>>>(out, ND);
    }
  }
  // else: workspace too small for any correct implementation; launch nothing.
}